// PairwiseMax_10926396801967
// MI455X (gfx1250) — compile-verified
//
#include <hip/hip_runtime.h>
#include <hip/hip_bf16.h>

// PairwiseMax: out[b, 0:256]   = x0[b,i] * (x0[b,i] >= 0 ? max_j x1[b,j] : min_j x1[b,j])
//              out[b, 256:384] = x2[b, 0:128]
// Pure bandwidth kernel: 16 MB total traffic -> ~0.7us at 23.3 TB/s.
// One wave32 per row; x1 row staged via CDNA5 async global->LDS copy.

#define AS1 __attribute__((address_space(1)))
#define AS3 __attribute__((address_space(3)))

typedef __attribute__((ext_vector_type(4))) int v4i;

__global__ __launch_bounds__(256) void PairwiseMax_kernel(
    const float* __restrict__ x0,
    const float* __restrict__ x1,
    const float* __restrict__ x2,
    float* __restrict__ out,
    int B)
{
    constexpr int D1 = 256, D2 = 256, F = 128, OW = D1 + F;  // 384
    __shared__ float s_x1[8 * D2];  // 8 rows * 1KB = 8KB

    const int tid  = threadIdx.x;
    const int wave = tid >> 5;
    const int lane = tid & 31;
    const int row  = blockIdx.x * 8 + wave;
    if (row >= B) return;  // whole-wave uniform; B=4096 -> never taken

    // ---- Stage x1 row into LDS with CDNA5 async copy (2 x b128 per lane) ----
    const float* g1 = x1 + (size_t)row * D2 + lane * 4;   // lane covers 16B, x2
    float*       l1 = &s_x1[wave * D2 + lane * 4];

#if __has_builtin(__builtin_amdgcn_global_load_async_to_lds_b128)
    __builtin_amdgcn_global_load_async_to_lds_b128(
        (AS1 v4i*)g1, (AS3 v4i*)l1, 0,   0);
    __builtin_amdgcn_global_load_async_to_lds_b128(
        (AS1 v4i*)g1, (AS3 v4i*)l1, 512, 0);   // imm offset applies to both addrs (ISA 08 §4.4)
#else
    {
        unsigned loff = (unsigned)(size_t)(AS3 float*)l1;  // LDS byte offset
        asm volatile(
            "global_load_async_to_lds_b128 %0, %1, off\n\t"
            "global_load_async_to_lds_b128 %0, %1, off offset:512"
            :: "v"(loff), "v"(g1) : "memory");
    }
#endif

#if __has_builtin(__builtin_amdgcn_s_wait_asynccnt)
    __builtin_amdgcn_s_wait_asynccnt(0);
#else
    asm volatile("s_wait_asynccnt 0" ::: "memory");
#endif
    asm volatile("" ::: "memory");  // keep LDS reads below the wait

    // ---- Row max/min: 8 elems/lane from LDS, then wave32 xor-shuffle tree ----
    const float* rowp = &s_x1[wave * D2];
    float4 a = *(const float4*)(rowp + lane * 4);
    float4 b = *(const float4*)(rowp + 128 + lane * 4);

    float mx = fmaxf(fmaxf(fmaxf(a.x, a.y), fmaxf(a.z, a.w)),
                     fmaxf(fmaxf(b.x, b.y), fmaxf(b.z, b.w)));
    float mn = fminf(fminf(fminf(a.x, a.y), fminf(a.z, a.w)),
                     fminf(fminf(b.x, b.y), fminf(b.z, b.w)));
    #pragma unroll
    for (int m = 16; m >= 1; m >>= 1) {
        mx = fmaxf(mx, __shfl_xor(mx, m, 32));
        mn = fminf(mn, __shfl_xor(mn, m, 32));
    }

    // ---- Elementwise select-multiply on x0 (2 x float4 per lane) ----
    const float* x0row = x0 + (size_t)row * D1;
    float*       orow  = out + (size_t)row * OW;

    float4 u = *(const float4*)(x0row + lane * 4);
    float4 v = *(const float4*)(x0row + 128 + lane * 4);
    float4 r0, r1;
    r0.x = (u.x >= 0.0f) ? u.x * mx : u.x * mn;
    r0.y = (u.y >= 0.0f) ? u.y * mx : u.y * mn;
    r0.z = (u.z >= 0.0f) ? u.z * mx : u.z * mn;
    r0.w = (u.w >= 0.0f) ? u.w * mx : u.w * mn;
    r1.x = (v.x >= 0.0f) ? v.x * mx : v.x * mn;
    r1.y = (v.y >= 0.0f) ? v.y * mx : v.y * mn;
    r1.z = (v.z >= 0.0f) ? v.z * mx : v.z * mn;
    r1.w = (v.w >= 0.0f) ? v.w * mx : v.w * mn;
    *(float4*)(orow + lane * 4)       = r0;
    *(float4*)(orow + 128 + lane * 4) = r1;

    // ---- Concat x2 (1 x float4 per lane covers 128 floats/row) ----
    float4 t = *(const float4*)(x2 + (size_t)row * F + lane * 4);
    *(float4*)(orow + D1 + lane * 4) = t;
}

extern "C" void kernel_launch(void* const* d_in, const int* in_sizes, int n_in,
                              void* d_out, int out_size, void* d_ws, size_t ws_size,
                              hipStream_t stream) {
    (void)n_in; (void)out_size; (void)d_ws; (void)ws_size;
    const float* x0 = (const float*)d_in[0];
    const float* x1 = (const float*)d_in[1];
    const float* x2 = (const float*)d_in[2];
    float* out = (float*)d_out;

    const int B = in_sizes[0] / 256;          // x0 is [B, 256]
    const int blocks = (B + 7) / 8;           // 8 rows (waves) per 256-thread block
    PairwiseMax_kernel<<<blocks, 256, 0, stream>>>(x0, x1, x2, out, B);
}